// LSTM_NET_90941637525935
// MI455X (gfx1250) — compile-verified
//
#include <hip/hip_runtime.h>
#include <hip/hip_bf16.h>

// Problem sizes (compile-time constants from the reference).
#define B_   32768
#define T_   200
#define I_   6
#define H_   6
#define TH_  1200     // T_*H_
#define FF_  128
#define OUT_ 12

typedef __attribute__((ext_vector_type(2))) float v2f;
typedef __attribute__((ext_vector_type(8))) float v8f;

// ---- overflow-safe fast activations (v_exp_f32 path) ----
__device__ __forceinline__ float fast_sigmoid(float x) {
    return 1.0f / (1.0f + __expf(-x));              // x->+inf: 1, x->-inf: 0
}
__device__ __forceinline__ float fast_tanh(float x) {
    float a = fabsf(x);
    float e = __expf(2.0f * a);                     // may be +inf for large |x|
    float r = 1.0f - 2.0f / (e + 1.0f);             // inf -> 1.0 (no NaN)
    return copysignf(r, x);
}

// ---------------------------------------------------------------------------
// Kernel 1: transpose W1 [128,1200] -> W1T [1200,128] so GEMM B-fragment
// loads are lane-coalesced. 153600 elements; negligible cost.
// ---------------------------------------------------------------------------
__global__ void __launch_bounds__(256) w1_transpose_kernel(
        const float* __restrict__ W1, float* __restrict__ W1T) {
    int idx = blockIdx.x * 256 + threadIdx.x;
    if (idx >= TH_ * FF_) return;
    int k = idx >> 7;          // 0..1199
    int f = idx & 127;         // 0..127
    W1T[idx] = W1[f * TH_ + k];
}

// ---------------------------------------------------------------------------
// Kernel 2: LSTM. One lane per batch element; weights broadcast from LDS.
// Writes hs to workspace as [B, T*H] f32. 24B/step contiguous per-lane
// streaming of x: each 128B line is consumed over ~5 consecutive t's.
// ---------------------------------------------------------------------------
__global__ void __launch_bounds__(256) lstm_kernel(
        const float* __restrict__ x,
        const float* __restrict__ W_ih, const float* __restrict__ W_hh,
        const float* __restrict__ b_ih, const float* __restrict__ b_hh,
        float* __restrict__ HS) {
    __shared__ float sWih[4 * H_ * I_];   // 144
    __shared__ float sWhh[4 * H_ * H_];   // 144
    __shared__ float sB[4 * H_];          // 24 (b_ih + b_hh fused)

    int tid = threadIdx.x;
    if (tid < 144) { sWih[tid] = W_ih[tid]; sWhh[tid] = W_hh[tid]; }
    if (tid < 24)  { sB[tid] = b_ih[tid] + b_hh[tid]; }
    __syncthreads();

    int b = blockIdx.x * 256 + tid;
    const float* xb = x + (size_t)b * (T_ * I_);
    float*       hb = HS + (size_t)b * TH_;

    float h[H_], c[H_];
#pragma unroll
    for (int j = 0; j < H_; ++j) { h[j] = 0.0f; c[j] = 0.0f; }

#pragma unroll 1
    for (int t = 0; t < T_; ++t) {
        float xv[I_];
#pragma unroll
        for (int i = 0; i < I_; ++i) xv[i] = xb[t * I_ + i];

        float g[4 * H_];
#pragma unroll
        for (int gi = 0; gi < 4 * H_; ++gi) {
            float acc = sB[gi];
#pragma unroll
            for (int i = 0; i < I_; ++i) acc += sWih[gi * I_ + i] * xv[i];
#pragma unroll
            for (int j = 0; j < H_; ++j) acc += sWhh[gi * H_ + j] * h[j];
            g[gi] = acc;
        }
        // PyTorch gate order: i, f, g, o
#pragma unroll
        for (int j = 0; j < H_; ++j) {
            float ig = fast_sigmoid(g[j]);
            float fg = fast_sigmoid(g[H_ + j]);
            float gg = fast_tanh  (g[2 * H_ + j]);
            float og = fast_sigmoid(g[3 * H_ + j]);
            float cj = fg * c[j] + ig * gg;
            c[j] = cj;
            float hj = og * fast_tanh(cj);
            h[j] = hj;
            hb[t * H_ + j] = hj;
        }
    }
}

// ---------------------------------------------------------------------------
// Kernel 3: fused MLP head with f32 WMMA (16x16x4).
//  Per wave: one 32-row batch tile (two 16-row M-subtiles) so the 16
//  B-fragment loads per k-step are amortized over 16 WMMAs (issue-bound ->
//  memory-bound; 157 MB hs stream ~= 6.7 us at 23.3 TB/s).
//   hid[32,128] = relu(HS_tile[32,1200] @ W1T + b1)   (300 k-steps x 16 WMMA)
//   out[32,12]  = hid @ W2T_padded + b2               (2 x 32 k-steps x WMMA)
// Fragment layouts (ISA 7.12.2, wave32):
//   A (16x4): lane l: M = l&15, K = 2*(l>>4) + reg
//   B (4x16): lane l: N = l&15, K = 2*(l>>4) + reg
//   C/D     : reg r:  M = r + 8*(l>>4), N = l&15
// ---------------------------------------------------------------------------
__global__ void __launch_bounds__(256) mlp_wmma_kernel(
        const float* __restrict__ HS, const float* __restrict__ W1T,
        const float* __restrict__ b1, const float* __restrict__ W2,
        const float* __restrict__ b2, float* __restrict__ out) {
    __shared__ float sW2T[FF_ * 16];        // [k][n], n padded 12->16 w/ zeros
    __shared__ float sHid[8 * 16 * 132];    // per-wave 16x132 staging buffer

    int tid = threadIdx.x;
    // Stage zero-padded W2^T once per block.
    for (int idx = tid; idx < FF_ * 16; idx += 256) {
        int k = idx >> 4, n = idx & 15;
        sW2T[idx] = (n < OUT_) ? W2[n * FF_ + k] : 0.0f;
    }
    __syncthreads();

    const int wave = tid >> 5;
    const int lane = tid & 31;
    const int lh   = lane & 15;   // N index / M index within half
    const int hh   = lane >> 4;   // which half-wave (K pair selector)

    const int mtile   = blockIdx.x * 8 + wave;   // 0..1023
    const int rowbase = mtile * 32;

    v8f z = {0.f, 0.f, 0.f, 0.f, 0.f, 0.f, 0.f, 0.f};
    v8f acc[16];                                  // [m-subtile][nt]
#pragma unroll
    for (int i = 0; i < 16; ++i) acc[i] = z;

    // -------- layer 1: [32 x 1200] @ [1200 x 128] --------
    const float* arow0 = HS + (size_t)(rowbase + lh) * TH_ + 2 * hh;
    const float* arow1 = arow0 + (size_t)16 * TH_;
#pragma unroll 2
    for (int k = 0; k < TH_; k += 4) {
        v2f a0, a1;
        a0.x = arow0[k];
        a0.y = arow0[k + 1];
        a1.x = arow1[k];
        a1.y = arow1[k + 1];
        const float* wk = W1T + (size_t)(k + 2 * hh) * FF_ + lh;
#pragma unroll
        for (int nt = 0; nt < 8; ++nt) {
            v2f bf;
            bf.x = wk[nt * 16];          // K = k + 2*hh
            bf.y = wk[FF_ + nt * 16];    // K = k + 2*hh + 1
            acc[nt] = __builtin_amdgcn_wmma_f32_16x16x4_f32(
                false, a0, false, bf, (short)0, acc[nt], false, false);
            acc[8 + nt] = __builtin_amdgcn_wmma_f32_16x16x4_f32(
                false, a1, false, bf, (short)0, acc[8 + nt], false, false);
        }
    }

    // b1 bias values for this lane's N columns.
    float b1v[8];
#pragma unroll
    for (int nt = 0; nt < 8; ++nt) b1v[nt] = b1[nt * 16 + lh];
    float bo = (lh < OUT_) ? b2[lh] : 0.0f;

    float* myHid = sHid + wave * (16 * 132);

    // -------- layer 2 (per 16-row subtile): [16 x 128] @ [128 x 16pad] -----
#pragma unroll 1
    for (int s = 0; s < 2; ++s) {
        // bias + relu, stage wave-private hid subtile to LDS (stride 132 to
        // spread the lh-row reads across banks).
#pragma unroll
        for (int nt = 0; nt < 8; ++nt) {
#pragma unroll
            for (int r = 0; r < 8; ++r) {
                int row = r + 8 * hh;
                float v = acc[s * 8 + nt][r] + b1v[nt];
                v = v > 0.0f ? v : 0.0f;
                myHid[row * 132 + nt * 16 + lh] = v;
            }
        }
        // LDS ops are in-order within a wave: the RAW below (and the WAR
        // with the next subtile's stores) is safe; compiler adds waits.
        v8f acc2 = z;
#pragma unroll
        for (int k2 = 0; k2 < FF_; k2 += 4) {
            v2f a2, bf2;
            a2.x  = myHid[lh * 132 + k2 + 2 * hh];
            a2.y  = myHid[lh * 132 + k2 + 2 * hh + 1];
            bf2.x = sW2T[(k2 + 2 * hh) * 16 + lh];
            bf2.y = sW2T[(k2 + 2 * hh + 1) * 16 + lh];
            acc2 = __builtin_amdgcn_wmma_f32_16x16x4_f32(
                false, a2, false, bf2, (short)0, acc2, false, false);
        }
#pragma unroll
        for (int r = 0; r < 8; ++r) {
            int row = rowbase + s * 16 + r + 8 * hh;
            if (lh < OUT_) out[row * OUT_ + lh] = acc2[r] + bo;
        }
    }
}

// ---------------------------------------------------------------------------
// Launch: W1 transpose -> LSTM -> fused WMMA MLP.
// Workspace: W1T (614,400 B) | HS [B,1200] f32 (157,286,400 B).
// ---------------------------------------------------------------------------
extern "C" void kernel_launch(void* const* d_in, const int* in_sizes, int n_in,
                              void* d_out, int out_size, void* d_ws, size_t ws_size,
                              hipStream_t stream) {
    const float* x    = (const float*)d_in[0];
    const float* W_ih = (const float*)d_in[1];
    const float* W_hh = (const float*)d_in[2];
    const float* b_ih = (const float*)d_in[3];
    const float* b_hh = (const float*)d_in[4];
    const float* W1   = (const float*)d_in[5];
    const float* b1   = (const float*)d_in[6];
    const float* W2   = (const float*)d_in[7];
    const float* b2   = (const float*)d_in[8];
    float* out = (float*)d_out;

    float* W1T = (float*)d_ws;
    size_t hs_off = ((size_t)TH_ * FF_ * sizeof(float) + 255) & ~(size_t)255;
    float* HS = (float*)((char*)d_ws + hs_off);

    w1_transpose_kernel<<<(TH_ * FF_ + 255) / 256, 256, 0, stream>>>(W1, W1T);
    lstm_kernel<<<B_ / 256, 256, 0, stream>>>(x, W_ih, W_hh, b_ih, b_hh, HS);
    // 128 blocks x 8 waves x 32 rows = 32768 batch rows.
    mlp_wmma_kernel<<<B_ / 256, 256, 0, stream>>>(HS, W1T, b1, W2, b2, out);
}